// Sinkhorn_46926812676491
// MI455X (gfx1250) — compile-verified
//
#include <hip/hip_runtime.h>
#include <math.h>

#define N_ORB 4096

typedef __attribute__((ext_vector_type(2))) float v2f;
typedef __attribute__((ext_vector_type(8))) float v8f;
typedef __attribute__((ext_vector_type(4))) unsigned int v4u;
typedef __attribute__((ext_vector_type(8))) int v8i;
typedef __attribute__((ext_vector_type(4))) int v4i;

// workspace layout (floats)
#define WS_MIN_EPS 0
#define WS_SUM_N   1
#define DK_OFF     16
#define F_OFF      (DK_OFF + N_ORB)       // F_M for M=0..N  (N+1 floats)
#define PRE_OFF    (F_OFF + N_ORB + 8)    // un-normalized new_eps (N floats)

// ---------------------------------------------------------------- WMMA helper
// D = A(16x4) * ones(4x16) + C : every column of D accumulates the row sum of A,
// in exact fp32 (V_WMMA_F32_16X16X4_F32).
__device__ __forceinline__ v8f wmma_rowsum_f32(v2f a, v8f c) {
  const v2f ones = {1.0f, 1.0f};
  return __builtin_amdgcn_wmma_f32_16x16x4_f32(false, a, false, ones,
                                               (short)0, c, false, false);
}

__device__ __forceinline__ float v8_get(v8f v, int i) {
  float r = v[0];
  if (i == 1) r = v[1];
  if (i == 2) r = v[2];
  if (i == 3) r = v[3];
  if (i == 4) r = v[4];
  if (i == 5) r = v[5];
  if (i == 6) r = v[6];
  if (i == 7) r = v[7];
  return r;
}

// ---------------------------------------------------------------- TDM helpers
#if __has_builtin(__builtin_amdgcn_tensor_load_to_lds)
#define HAVE_TDM 1
typedef __attribute__((address_space(3))) float lds_f32;

__device__ __forceinline__ unsigned lds_off_of(float* p) {
  return (unsigned)(uintptr_t)(lds_f32*)p;   // as3 ptrtoint == LDS byte offset
}

// 1D D# descriptor: contiguous copy of `elems` f32 between global and LDS.
// Group0: count=1, lds_addr, global_addr[56:0], type=2 ("image").
// Group1: data_size=4B, tensor_dim0=tile_dim0=elems, tensor_dim1=1,
//         tensor_dim0_stride=elems; tile_dim1..2 unused. Groups 2/3 zero
// (tile_dim3/4 unused). This toolchain's builtin takes 6 args:
//   (v4u g0, v8i g1, v4i g2, v4i g3, v8i extra, i32 cpol)
__device__ __forceinline__ void tdm_desc_1d(v4u& g0, v8i& g1,
                                            unsigned lds_off, const void* gptr,
                                            unsigned elems) {
  const unsigned long long ga = (unsigned long long)(uintptr_t)gptr;
  g0[0] = 1u;                                          // count=1 (user mode)
  g0[1] = lds_off;                                     // lds_addr (bytes)
  g0[2] = (unsigned)(ga & 0xFFFFFFFFu);                // global_addr[31:0]
  g0[3] = ((unsigned)(ga >> 32) & 0x1FFFFFFu) | (2u << 30);  // [56:32] | type=2
  g1[0] = 0x20000;                                     // data_size=2 (4 bytes)
  g1[1] = (int)((elems & 0xFFFFu) << 16);              // tensor_dim0[15:0] @63:48
  g1[2] = (int)((elems >> 16) | (1u << 16));           // dim0[31:16] | dim1=1
  g1[3] = (int)((elems & 0xFFFFu) << 16);              // dim1 hi=0 | tile_dim0
  g1[4] = 0;                                           // tile_dim1/2 unused
  g1[5] = (int)elems;                                  // dim0_stride[31:0]
  g1[6] = 0;                                           // stride0 hi | stride1 lo
  g1[7] = 0;
}

__device__ __forceinline__ void tdm_load_1d(float* lds, const void* gsrc,
                                            unsigned elems) {
  v4u g0; v8i g1;
  const v4i z4 = {0, 0, 0, 0};
  const v8i z8 = {0, 0, 0, 0, 0, 0, 0, 0};
  tdm_desc_1d(g0, g1, lds_off_of(lds), gsrc, elems);
  __builtin_amdgcn_tensor_load_to_lds(g0, g1, z4, z4, z8, 0);
}

__device__ __forceinline__ void tdm_store_1d(float* lds, void* gdst,
                                             unsigned elems) {
  v4u g0; v8i g1;
  const v4i z4 = {0, 0, 0, 0};
  const v8i z8 = {0, 0, 0, 0, 0, 0, 0, 0};
  tdm_desc_1d(g0, g1, lds_off_of(lds), gdst, elems);
  __builtin_amdgcn_tensor_store_from_lds(g0, g1, z4, z4, z8, 0);
}
#else
#define HAVE_TDM 0
#endif

// stage `elems` floats global->LDS: wave 0 drives the TDM, everyone barriers
__device__ __forceinline__ void stage_to_lds(float* lds, const float* g,
                                             int elems, int tid, int nthreads) {
#if HAVE_TDM
  (void)nthreads;
  if (tid < 32) {
    tdm_load_1d(lds, g, (unsigned)elems);
    __builtin_amdgcn_s_wait_tensorcnt(0);
  }
#else
  for (int j = tid; j < elems; j += nthreads) lds[j] = g[j];
#endif
}

// ---------------- K0: min(eps), sum(n) ----------------
__global__ void __launch_bounds__(1024)
k0_scalars(const float* __restrict__ eps, const float* __restrict__ n,
           float* __restrict__ ws) {
  __shared__ float mw[32], sw[32];
  const int tid = threadIdx.x;
  float lm = __builtin_inff();
  float ls = 0.f;
  for (int j = tid; j < N_ORB; j += 1024) { lm = fminf(lm, eps[j]); ls += n[j]; }
  for (int off = 16; off > 0; off >>= 1) {
    lm = fminf(lm, __shfl_xor(lm, off, 32));
    ls += __shfl_xor(ls, off, 32);
  }
  const int w = tid >> 5, lane = tid & 31;
  if (lane == 0) { mw[w] = lm; sw[w] = ls; }
  __syncthreads();
  if (w == 0) {
    lm = mw[lane]; ls = sw[lane];
    for (int off = 16; off > 0; off >>= 1) {
      lm = fminf(lm, __shfl_xor(lm, off, 32));
      ls += __shfl_xor(ls, off, 32);
    }
    if (lane == 0) { ws[WS_MIN_EPS] = lm; ws[WS_SUM_N] = ls; }
  }
}

// ---------------- K1: Dk[k-1] = k*min_eps - log(S_k)/beta,
//                  S_k = sum_p exp(-beta*k*(eps[p]-min_eps))  (WMMA row sums) ----
__global__ void __launch_bounds__(256)
k1_single_particle(const float* __restrict__ eps, const float* __restrict__ beta_p,
                   float* __restrict__ ws) {
  __shared__ float epsL[N_ORB];
  const int tid = threadIdx.x;
  stage_to_lds(epsL, eps, N_ORB, tid, 256);
  __syncthreads();

  const float beta = beta_p[0];
  const float me   = ws[WS_MIN_EPS];
  const int w = tid >> 5, lane = tid & 31;
  const int gw  = blockIdx.x * 8 + w;   // global wave id: owns 16 consecutive k
  const int row = lane & 15;            // WMMA A-row = k index within wave
  const int h   = lane >> 4;            // K-slice half (lanes 16..31 give K=2,3)
  const float kk = (float)(gw * 16 + row + 1);
  const float bk = beta * kk;

  v8f acc = {};
  for (int p0 = 0; p0 < N_ORB; p0 += 4) {
    v2f a;
    a.x = __expf(-bk * (epsL[p0 + 2 * h]     - me));
    a.y = __expf(-bk * (epsL[p0 + 2 * h + 1] - me));
    acc = wmma_rowsum_f32(a, acc);   // args <= 0, terms <= 1: overflow-free
  }

  // rows 0..7 live in lanes 0..15 (VGPR r), rows 8..15 in lanes 16..31
  int r = -1;
  if (lane < 8)                      r = lane;
  else if (lane >= 16 && lane < 24)  r = 8 + (lane - 16);
  if (r >= 0) {
    const float S = v8_get(acc, (lane < 8) ? lane : (lane - 16));
    const int k0b = gw * 16 + r;                  // 0-based slot for k = k0b+1
    ws[DK_OFF + k0b] = (float)(k0b + 1) * me - __logf(S) / beta;
  }
}

// ---------------- K2: serial bosonic recursion, one workgroup ----------------
// G[i] = min_t e_t + (log i - log sum_t exp(-beta*(e_t-min)))/beta,
//   e_t = Dk[t] + G[i-1-t], t=0..i-1
__global__ void __launch_bounds__(1024)
k2_recursion(const float* __restrict__ beta_p, float* __restrict__ ws) {
  __shared__ float DkL[N_ORB];
  __shared__ float GL[N_ORB + 1];
  __shared__ float mw[32], sw[32];
  const int tid = threadIdx.x;
  const float beta = beta_p[0];
  stage_to_lds(DkL, ws + DK_OFF, N_ORB, tid, 1024);
  if (tid == 0) GL[0] = 0.f;
  __syncthreads();

  const int w = tid >> 5, lane = tid & 31;
  for (int i = 1; i <= N_ORB; ++i) {
    // exact per-thread (min, sum) logsumexp partial
    float lm = __builtin_inff();
    for (int t = tid; t < i; t += 1024) lm = fminf(lm, DkL[t] + GL[i - 1 - t]);
    float ls = 0.f;
    for (int t = tid; t < i; t += 1024)
      ls += __expf(-beta * (DkL[t] + GL[i - 1 - t] - lm));
    // wave32 pair-merge
    for (int off = 16; off > 0; off >>= 1) {
      const float m2 = __shfl_xor(lm, off, 32);
      const float s2 = __shfl_xor(ls, off, 32);
      const float mn = fminf(lm, m2);
      const float t1 = (ls == 0.f) ? 0.f : ls * __expf(-beta * (lm - mn));
      const float t2 = (s2 == 0.f) ? 0.f : s2 * __expf(-beta * (m2 - mn));
      lm = mn; ls = t1 + t2;
    }
    if (lane == 0) { mw[w] = lm; sw[w] = ls; }
    __syncthreads();
    if (w == 0) {
      lm = mw[lane]; ls = sw[lane];
      for (int off = 16; off > 0; off >>= 1) {
        const float m2 = __shfl_xor(lm, off, 32);
        const float s2 = __shfl_xor(ls, off, 32);
        const float mn = fminf(lm, m2);
        const float t1 = (ls == 0.f) ? 0.f : ls * __expf(-beta * (lm - mn));
        const float t2 = (s2 == 0.f) ? 0.f : s2 * __expf(-beta * (m2 - mn));
        lm = mn; ls = t1 + t2;
      }
      if (lane == 0)
        GL[i] = lm + (__logf((float)i) - __logf(ls)) / beta;
    }
    __syncthreads();
  }
#if HAVE_TDM
  if (tid < 32) {
    tdm_store_1d(GL, ws + F_OFF, N_ORB + 1);
    __builtin_amdgcn_s_wait_tensorcnt(0);
  }
#else
  for (int j = tid; j <= N_ORB; j += 1024) ws[F_OFF + j] = GL[j];
#endif
}

// ---------------- K3: aux free energies + eps update (WMMA row sums) --------
// E0[p,k] = G[N-1-k] + k*eps[p]  (k=0..N-1) ; E1[p,k] = G[N-k] + k*eps[p] (k=0..N)
__global__ void __launch_bounds__(256)
k3_aux(const float* __restrict__ eps, const float* __restrict__ n,
       const float* __restrict__ beta_p, float* __restrict__ ws) {
  __shared__ float GL[N_ORB + 1];
  __shared__ float sum0[8][16];
  __shared__ float sum1[8][16];
  const int tid = threadIdx.x;
  stage_to_lds(GL, ws + F_OFF, N_ORB + 1, tid, 256);
  __syncthreads();

  const float beta = beta_p[0];
  const int w = tid >> 5, lane = tid & 31;
  const int h = lane >> 4;
  const int pp = (blockIdx.x * 8 + w) * 16 + (lane & 15);  // WMMA A-row = orbital
  const float ep = eps[pp];

  // pass 1: per-orbital min energies (two lanes split the k range)
  float m0 = __builtin_inff(), m1 = __builtin_inff();
  for (int k = h; k < N_ORB; k += 2) {
    const float t = (float)k * ep;
    m0 = fminf(m0, GL[N_ORB - 1 - k] + t);
    m1 = fminf(m1, GL[N_ORB - k] + t);
  }
  if (h == 0) m1 = fminf(m1, GL[0] + (float)N_ORB * ep);  // E1's k=N term
  m0 = fminf(m0, __shfl_xor(m0, 16, 32));
  m1 = fminf(m1, __shfl_xor(m1, 16, 32));

  // pass 2: shifted exp terms accumulated by the matrix pipe (exact fp32)
  v8f a0acc = {}, a1acc = {};
  for (int kc = 0; kc < N_ORB; kc += 4) {
    const int ka = kc + 2 * h;
    const float ta = (float)ka * ep, tb = ta + ep;
    v2f a0, a1;
    a0.x = __expf(-beta * (GL[N_ORB - 1 - ka] + ta - m0));
    a0.y = __expf(-beta * (GL[N_ORB - 2 - ka] + tb - m0));
    a1.x = __expf(-beta * (GL[N_ORB - ka]     + ta - m1));
    a1.y = __expf(-beta * (GL[N_ORB - 1 - ka] + tb - m1));
    a0acc = wmma_rowsum_f32(a0, a0acc);
    a1acc = wmma_rowsum_f32(a1, a1acc);
  }

  // row extraction: lane 0 holds rows 0..7, lane 16 holds rows 8..15
  if (lane == 0) {
#pragma unroll
    for (int r = 0; r < 8; ++r) { sum0[w][r] = a0acc[r]; sum1[w][r] = a1acc[r]; }
  }
  if (lane == 16) {
#pragma unroll
    for (int r = 0; r < 8; ++r) { sum0[w][8 + r] = a0acc[r]; sum1[w][8 + r] = a1acc[r]; }
  }
  __syncthreads();

  if (lane < 16) {
    const float S0 = sum0[w][lane];
    float S1 = sum1[w][lane] + __expf(-beta * (GL[0] + (float)N_ORB * ep - m1));
    const float Fp0 = m0 - __logf(S0) / beta;
    const float Fp1 = m1 - __logf(S1) / beta;
    const float nn  = n[pp] * ((float)N_ORB / ws[WS_SUM_N]);
    ws[PRE_OFF + pp] = -__logf(nn / (1.f + nn)) / beta + Fp1 - Fp0;
  }
}

// ---------------- K4: subtract occupation-weighted mean ----------------
__global__ void __launch_bounds__(1024)
k4_normalize(const float* __restrict__ n_in, const float* __restrict__ ws,
             float* __restrict__ out) {
  __shared__ float sw[32];
  __shared__ float meanL;
  const int tid = threadIdx.x;
  float ls = 0.f;
  for (int j = tid; j < N_ORB; j += 1024) ls += n_in[j] * ws[PRE_OFF + j];
  for (int off = 16; off > 0; off >>= 1) ls += __shfl_xor(ls, off, 32);
  const int w = tid >> 5, lane = tid & 31;
  if (lane == 0) sw[w] = ls;
  __syncthreads();
  if (tid == 0) {
    float tot = 0.f;
    for (int q = 0; q < 32; ++q) tot += sw[q];
    meanL = tot / ws[WS_SUM_N];   // weighted mean is scale-invariant in n
  }
  __syncthreads();
  const float mean = meanL;
  for (int j = tid; j < N_ORB; j += 1024) out[j] = ws[PRE_OFF + j] - mean;
}

extern "C" void kernel_launch(void* const* d_in, const int* in_sizes, int n_in,
                              void* d_out, int out_size, void* d_ws, size_t ws_size,
                              hipStream_t stream) {
  (void)in_sizes; (void)n_in; (void)out_size; (void)ws_size;
  const float* eps  = (const float*)d_in[0];
  const float* n    = (const float*)d_in[1];
  const float* beta = (const float*)d_in[2];
  float* ws  = (float*)d_ws;
  float* out = (float*)d_out;

  k0_scalars<<<1, 1024, 0, stream>>>(eps, n, ws);
  k1_single_particle<<<32, 256, 0, stream>>>(eps, beta, ws);
  k2_recursion<<<1, 1024, 0, stream>>>(beta, ws);
  k3_aux<<<32, 256, 0, stream>>>(eps, n, beta, ws);
  k4_normalize<<<1, 1024, 0, stream>>>(n, ws, out);
}